// EGCN_NoStop_84121229460224
// MI455X (gfx1250) — compile-verified
//
#include <hip/hip_runtime.h>
#include <hip/hip_bf16.h>

typedef __attribute__((ext_vector_type(2))) float v2f;
typedef __attribute__((ext_vector_type(8))) float v8f;

#define EPS 1e-5f
#define H 128

// ---------------- degree / dinv ----------------
__global__ void deg_kernel(const int* __restrict__ dst, float* __restrict__ deg, int e) {
    int i = blockIdx.x * blockDim.x + threadIdx.x;
    if (i < e) atomicAdd(&deg[dst[i]], 1.0f);
}

__global__ void dinv_kernel(float* __restrict__ dinv, int n) {
    int i = blockIdx.x * blockDim.x + threadIdx.x;
    if (i < n) dinv[i] = rsqrtf(fmaxf(dinv[i], 1.0f));
}

// ---------------- SpMM: agg[dst] += dinv[src] * x[src] (one wave per edge, 4 floats/lane) --
__global__ void spmm_edge(const float* __restrict__ x, const int* __restrict__ src,
                          const int* __restrict__ dst, const float* __restrict__ dinv,
                          float* __restrict__ agg, int e) {
    int gid  = blockIdx.x * blockDim.x + threadIdx.x;
    int edge = gid >> 5;
    int lane = gid & 31;
    if (edge >= e) return;
    int s = src[edge];
    int d = dst[edge];
    float w = dinv[s];
    const float4 xv = *(const float4*)(x + (size_t)s * H + lane * 4);
    float* o = agg + (size_t)d * H + lane * 4;
    atomicAdd(o + 0, w * xv.x);
    atomicAdd(o + 1, w * xv.y);
    atomicAdd(o + 2, w * xv.z);
    atomicAdd(o + 3, w * xv.w);
}

// ------- combine (float4): io = alpha * io * dinv[row] + beta * prev (in place) ----------
__global__ void combine_scale4(float4* __restrict__ io, const float* __restrict__ dinv,
                               const float4* __restrict__ prev, float alpha, float beta,
                               long long total4) {
    long long i = (long long)blockIdx.x * blockDim.x + threadIdx.x;
    if (i >= total4) return;
    const float w = alpha * dinv[i >> 5];       // 32 float4 per row (H == 128)
    float4 v = io[i];
    v.x *= w; v.y *= w; v.z *= w; v.w *= w;
    if (prev) {
        float4 p = prev[i];
        v.x += beta * p.x; v.y += beta * p.y; v.z += beta * p.z; v.w += beta * p.w;
    }
    io[i] = v;
}

// ---------------- WMMA GEMM core: acc += A(16x128 rows of X) @ W-term -------------------
template <int DOUT>
__device__ __forceinline__ v8f gemm_term(v8f acc, const float* __restrict__ arow,
                                         const float* __restrict__ Wt, int ncol, int half) {
#pragma unroll 8
    for (int k = 0; k < H; k += 4) {
        const int ks = k + half * 2;
        const float2 av = *(const float2*)(arow + ks);   // global_load_b64
        v2f a; a.x = av.x; a.y = av.y;
        const float* __restrict__ wp = Wt + ks * DOUT + ncol;
        v2f b; b.x = wp[0]; b.y = wp[DOUT];
        acc = __builtin_amdgcn_wmma_f32_16x16x4_f32(
            false, a, false, b, (short)0, acc, false, false);
    }
    return acc;
}

// out = relu?( sum_t X_t @ W_t + b ) [+ residual]; one wave -> one 16x16 fp32 tile
template <int TERMS, int DOUT>
__global__ void cheb_gemm_wmma(const float* __restrict__ X0, const float* __restrict__ X1,
                               const float* __restrict__ X2, const float* __restrict__ W,
                               const float* __restrict__ bias, const float* __restrict__ residual,
                               float* __restrict__ out, int n, int relu) {
    const int lane = threadIdx.x & 31;
    const int wave = threadIdx.x >> 5;
    constexpr int wavesPerRow  = DOUT / 16;     // 8 (dout=128) or 4 (dout=64)
    constexpr int rowsPerBlock = 8 / wavesPerRow;
    const int rowTile = blockIdx.x * rowsPerBlock + wave / wavesPerRow;
    const int m0 = rowTile * 16;
    const int n0 = (wave % wavesPerRow) * 16;
    if (m0 >= n) return;                        // wave-uniform exit (EXEC all-1s for WMMA)

    const int half = lane >> 4;                 // 0: K=k,k+1 ; 1: K=k+2,k+3
    const int l16  = lane & 15;
    const int mr   = (m0 + l16) < n ? (m0 + l16) : (n - 1);
    const size_t rowoff = (size_t)mr * H;
    const int ncol = n0 + l16;

    v8f acc = {};
    acc = gemm_term<DOUT>(acc, X0 + rowoff, W, ncol, half);
    if constexpr (TERMS > 1) acc = gemm_term<DOUT>(acc, X1 + rowoff, W + 1 * H * DOUT, ncol, half);
    if constexpr (TERMS > 2) acc = gemm_term<DOUT>(acc, X2 + rowoff, W + 2 * H * DOUT, ncol, half);

    const float bv = bias ? bias[ncol] : 0.0f;
#pragma unroll
    for (int j = 0; j < 8; ++j) {
        const int m = m0 + j + half * 8;        // C/D: VGPR j -> M = j (+8 for lanes 16-31)
        if (m < n) {
            float v = acc[j] + bv;
            if (relu) v = fmaxf(v, 0.0f);
            if (residual) v += residual[(size_t)m * DOUT + ncol];
            out[(size_t)m * DOUT + ncol] = v;
        }
    }
}

// ---------------- BatchNorm (training-mode batch stats over rows) -------------------------
__global__ void bn_partial(const float* __restrict__ x, float* __restrict__ sums,
                           float* __restrict__ sumsq, int n) {
    __shared__ float s1[256];
    __shared__ float s2[256];
    const int col = threadIdx.x & (H - 1);
    const int rh  = threadIdx.x >> 7;           // 0 or 1
    const int r0  = blockIdx.x * 128;
    const int r1  = (r0 + 128 < n) ? (r0 + 128) : n;
    float s = 0.0f, q = 0.0f;
    for (int r = r0 + rh; r < r1; r += 2) {
        float v = x[(size_t)r * H + col];
        s += v;
        q += v * v;
    }
    s1[threadIdx.x] = s;
    s2[threadIdx.x] = q;
    __syncthreads();
    if (rh == 0) {
        s += s1[threadIdx.x + 128];
        q += s2[threadIdx.x + 128];
        atomicAdd(&sums[col], s);
        atomicAdd(&sumsq[col], q);
    }
}

__global__ void bn_finalize(const float* __restrict__ sums, const float* __restrict__ sumsq,
                            const float* __restrict__ gamma, const float* __restrict__ beta,
                            float* __restrict__ scale, float* __restrict__ shift, int n) {
    int c = threadIdx.x;
    if (c < H) {
        float inv_n = 1.0f / (float)n;
        float mu  = sums[c] * inv_n;
        float var = sumsq[c] * inv_n - mu * mu;
        float a = gamma[c] * rsqrtf(var + EPS);
        scale[c] = a;
        shift[c] = beta[c] - mu * a;
    }
}

__global__ void bn_apply4(float4* __restrict__ x, const float4* __restrict__ scale,
                          const float4* __restrict__ shift, long long total4) {
    long long i = (long long)blockIdx.x * blockDim.x + threadIdx.x;
    if (i >= total4) return;
    const int c = (int)(i & 31);                // 32 float4 per row
    const float4 a = scale[c];
    const float4 b = shift[c];
    float4 v = x[i];
    v.x = fmaf(v.x, a.x, b.x);
    v.y = fmaf(v.y, a.y, b.y);
    v.z = fmaf(v.z, a.z, b.z);
    v.w = fmaf(v.w, a.w, b.w);
    x[i] = v;
}

// =========================================================================================
extern "C" void kernel_launch(void* const* d_in, const int* in_sizes, int n_in,
                              void* d_out, int out_size, void* d_ws, size_t ws_size,
                              hipStream_t stream) {
    const float* feat  = (const float*)d_in[0];
    const int*   src   = (const int*)d_in[1];
    const int*   dst   = (const int*)d_in[2];
    const float* W1    = (const float*)d_in[3];
    const float* b1    = (const float*)d_in[4];
    const float* W2    = (const float*)d_in[5];
    const float* b2    = (const float*)d_in[6];
    const float* W3    = (const float*)d_in[7];
    const float* b3    = (const float*)d_in[8];
    const float* gamma = (const float*)d_in[9];
    const float* beta  = (const float*)d_in[10];
    const float* mw1   = (const float*)d_in[11];
    const float* mb1   = (const float*)d_in[12];
    const float* mw2   = (const float*)d_in[13];
    const float* mb2   = (const float*)d_in[14];

    const int n = in_sizes[0] / H;
    const int e = in_sizes[1];

    float* ws    = (float*)d_ws;
    float* dinv  = ws;
    float* sums  = ws + n;
    float* sumsq = sums + H;
    float* scale = sumsq + H;
    float* shift = scale + H;
    size_t ofs   = ((size_t)n + 4 * H + 255) & ~(size_t)255;
    const size_t big = (size_t)n * H;
    float* P0 = ws + ofs;
    float* P1 = P0 + big;
    float* P2 = P1 + big;
    float* P3 = P2 + big;

    const long long total4 = (long long)big / 4;
    const int gridElem4 = (int)((total4 + 255) / 256);
    const int gridEdge  = (int)(((long long)e * 32 + 255) / 256);
    const int rowTiles  = (n + 15) / 16;

    // degrees -> dinv (in place)
    hipMemsetAsync(dinv, 0, (size_t)n * sizeof(float), stream);
    deg_kernel<<<(e + 255) / 256, 256, 0, stream>>>(dst, dinv, e);
    dinv_kernel<<<(n + 255) / 256, 256, 0, stream>>>(dinv, n);

    auto spmm = [&](const float* xin, float* agg) {
        hipMemsetAsync(agg, 0, big * sizeof(float), stream);
        spmm_edge<<<gridEdge, 256, 0, stream>>>(xin, src, dst, dinv, agg, e);
    };
    auto cheb = [&](const float* cur, const float* W, const float* b,
                    float* X1b, float* X2b, float* out, const float* resid) {
        spmm(cur, X1b);  // X1 = -spmm(cur)
        combine_scale4<<<gridElem4, 256, 0, stream>>>((float4*)X1b, dinv, nullptr,
                                                      -1.0f, 0.0f, total4);
        spmm(X1b, X2b);  // X2 = -2*spmm(X1) - cur
        combine_scale4<<<gridElem4, 256, 0, stream>>>((float4*)X2b, dinv, (const float4*)cur,
                                                      -2.0f, -1.0f, total4);
        cheb_gemm_wmma<3, H><<<rowTiles, 256, 0, stream>>>(cur, X1b, X2b, W, b, resid,
                                                           out, n, 1);
    };

    // cheb1 + relu
    cheb(feat, W1, b1, P0, P1, P2, nullptr);
    // batchnorm (in place on P2)
    hipMemsetAsync(sums, 0, 2 * H * sizeof(float), stream);
    bn_partial<<<(n + 127) / 128, 256, 0, stream>>>(P2, sums, sumsq, n);
    bn_finalize<<<1, H, 0, stream>>>(sums, sumsq, gamma, beta, scale, shift, n);
    bn_apply4<<<gridElem4, 256, 0, stream>>>((float4*)P2, (const float4*)scale,
                                             (const float4*)shift, total4);
    // cheb2 applied twice (shared weights) + relu
    cheb(P2, W2, b2, P0, P1, P3, nullptr);
    cheb(P3, W2, b2, P0, P1, P2, nullptr);
    // cheb3 + relu + residual (residual = input of cheb3)
    cheb(P2, W3, b3, P0, P1, P3, P2);
    // MLP head: relu(x @ mw1 + mb1) @ mw2 + mb2
    cheb_gemm_wmma<1, H><<<rowTiles, 256, 0, stream>>>(P3, nullptr, nullptr, mw1, mb1,
                                                       nullptr, P0, n, 1);
    cheb_gemm_wmma<1, 64><<<(rowTiles + 1) / 2, 256, 0, stream>>>(P0, nullptr, nullptr,
                                                                  mw2, mb2, nullptr,
                                                                  (float*)d_out, n, 0);
    (void)n_in; (void)out_size; (void)ws_size;
}